// EdgeFlexibleAttention_88493506166791
// MI455X (gfx1250) — compile-verified
//
#include <hip/hip_runtime.h>
#include <hip/hip_bf16.h>

// CDNA5 / gfx1250 implementation of EdgeFlexibleAttention.
//
// Precision: reference is fp32; total FLOPs (~16.4 GF) are tiny vs the matrix
// pipe, so we use V_WMMA_F32_16X16X4_F32 (full fp32 WMMA) rather than a lossy
// f16 downcast. Memory: Q+K+Y (~51 MB) are resident in the 192 MB L2, so the
// per-edge gathers are L2 hits; streaming traffic at 23.3 TB/s is tens of us.
// The reference's global max-subtraction cancels exactly in
// alpha = exp(s-m)/sum exp(s-m), and |score| <= 8, so it is dropped (identical
// math, exp() stays in range).
//
// CDNA5-specific paths exercised:
//   - v_wmma_f32_16x16x4_f32 (node GEMMs + per-edge e@We GEMM)
//   - global_load_async_to_lds_b128 + s_wait_asynccnt (We staging, ASYNCcnt)
//   - global_prefetch_b8 (L2 warm for We)

typedef __attribute__((ext_vector_type(2))) float v2f;
typedef __attribute__((ext_vector_type(8))) float v8f;
typedef __attribute__((ext_vector_type(4))) int   v4i;   // b128 payload type

#define DIMC 128
#define ADIM 64

#define AS1 __attribute__((address_space(1)))
#define AS3 __attribute__((address_space(3)))

#if __has_builtin(__builtin_amdgcn_global_load_async_to_lds_b128)
#define USE_ASYNC_LDS 1
#else
#define USE_ASYNC_LDS 0
#endif

__device__ __forceinline__ void wait_asynccnt0()
{
#if __has_builtin(__builtin_amdgcn_s_wait_asynccnt)
    __builtin_amdgcn_s_wait_asynccnt(0);
#else
    asm volatile("s_wait_asynccnt 0" ::: "memory");
#endif
}

// ---------------------------------------------------------------------------
// Kernel 1: node GEMMs.  Q = X@Wq [N,64], K = X@Wk [N,64], Y = X@Wy_w + b [N,128].
// One wave32 computes one 16x16 output tile, K-loop over 128 in steps of 4
// with V_WMMA_F32_16X16X4_F32.
// Wave-job index = rowTile*16 + sel;  sel 0..3 -> Q, 4..7 -> K, 8..15 -> Y.
// ---------------------------------------------------------------------------
__global__ void node_gemm_kernel(const float* __restrict__ X,
                                 const float* __restrict__ Wq,
                                 const float* __restrict__ Wk,
                                 const float* __restrict__ Wy_w,
                                 const float* __restrict__ Wy_b,
                                 float* __restrict__ Qo,
                                 float* __restrict__ Ko,
                                 float* __restrict__ Yo,
                                 int nNodes, int nRowTiles)
{
    const int wave = threadIdx.x >> 5;
    const int lane = threadIdx.x & 31;
    const int job  = blockIdx.x * 8 + wave;          // 8 waves per block
    const int rowTile = job >> 4;
    const int sel     = job & 15;
    if (rowTile >= nRowTiles) return;                // wave-uniform exit

    const float* W;
    float*       Out;
    int          n0, ncols;
    bool         addBias = false;
    if (sel < 4)       { W = Wq;   Out = Qo; ncols = ADIM; n0 = sel * 16; }
    else if (sel < 8)  { W = Wk;   Out = Ko; ncols = ADIM; n0 = (sel - 4) * 16; }
    else               { W = Wy_w; Out = Yo; ncols = DIMC; n0 = (sel - 8) * 16; addBias = true; }

    const int m0 = rowTile * 16;
    const int mr = lane & 15;       // row within tile (A) / col within tile (B,C)
    const int kh = lane >> 4;       // K half: lanes 0-15 -> K={0,1}, 16-31 -> K={2,3}
    const int col = n0 + mr;

    // Hoist bias out of the store loop (depends only on col) -> branchless epilogue.
    const float bias = addBias ? Wy_b[col] : 0.0f;

    int arow = m0 + mr;
    if (arow >= nNodes) arow = nNodes - 1;           // defensive clamp (N%16==0 here)
    const float* xrow = X + (size_t)arow * DIMC;

    v8f c = {};
    for (int k0 = 0; k0 < DIMC; k0 += 4) {
        v2f a, b;
        a.x = xrow[k0 + 2 * kh];
        a.y = xrow[k0 + 2 * kh + 1];
        b.x = W[(size_t)(k0 + 2 * kh)     * ncols + col];
        b.y = W[(size_t)(k0 + 2 * kh + 1) * ncols + col];
        // D = A(16x4) * B(4x16) + C   (fp32 WMMA)
        c = __builtin_amdgcn_wmma_f32_16x16x4_f32(false, a, false, b,
                                                  (short)0, c, false, false);
    }

    #pragma unroll
    for (int r = 0; r < 8; ++r) {
        const int row = m0 + r + 8 * kh;             // C layout: VGPR r -> M = r + 8*(lane/16)
        if (row < nNodes)
            Out[(size_t)row * ncols + col] = c[r] + bias;
    }
}

// ---------------------------------------------------------------------------
// Kernel 2: zero the atomic accumulation buffers (out [N,128] and denom [N]).
// ---------------------------------------------------------------------------
__global__ void zero_kernel(float* __restrict__ out, float* __restrict__ denom,
                            int nOut, int nDen)
{
    const int i = blockIdx.x * blockDim.x + threadIdx.x;
    if (i < nOut) out[i] = 0.0f;
    if (i < nDen) denom[i] = 0.0f;
}

// ---------------------------------------------------------------------------
// Kernel 3: per-edge score.  exps[e] = exp(Wphi . tanh(Q[src]+K[dst])),
// denom[src] += exps[e].  Q/K gathers hit L2 (each buffer is 12.8 MB).
// ---------------------------------------------------------------------------
__global__ void edge_score_kernel(const int* __restrict__ srcIdx,
                                  const int* __restrict__ dstIdx,
                                  const float* __restrict__ Q,
                                  const float* __restrict__ K,
                                  const float* __restrict__ Wphi,
                                  float* __restrict__ exps,
                                  float* __restrict__ denom,
                                  int nEdges)
{
    __shared__ float sPhi[ADIM];
    if (threadIdx.x < ADIM) sPhi[threadIdx.x] = Wphi[threadIdx.x];
    __syncthreads();

    const int e = blockIdx.x * blockDim.x + threadIdx.x;
    if (e >= nEdges) return;

    const int s = srcIdx[e];
    const int d = dstIdx[e];
    const float* q = Q + (size_t)s * ADIM;
    const float* k = K + (size_t)d * ADIM;

    float acc = 0.0f;
    #pragma unroll 8
    for (int i = 0; i < ADIM; ++i)
        acc += sPhi[i] * tanhf(q[i] + k[i]);

    const float ex = __expf(acc);                    // |acc| <= 8, always safe
    exps[e] = ex;
    atomicAdd(&denom[s], ex);
}

// ---------------------------------------------------------------------------
// Kernel 4: edge messages.  Per block: 128 edges, 8 waves.
//  - stage We [64x128] in LDS via GLOBAL_LOAD_ASYNC_TO_LDS_B128 (ASYNCcnt)
//  - stage e = tanh(Q[src]+K[dst]) [128x64] in LDS
//  - each wave: 16-edge x 128-col GEMM via 8 fp32 WMMA accumulators, K-loop 64
//  - epilogue: out[src] += alpha * (Y[dst] + ve)   (float atomics)
// ---------------------------------------------------------------------------
__global__ void edge_msg_kernel(const int* __restrict__ srcIdx,
                                const int* __restrict__ dstIdx,
                                const float* __restrict__ Q,
                                const float* __restrict__ K,
                                const float* __restrict__ Y,
                                const float* __restrict__ We,
                                const float* __restrict__ exps,
                                const float* __restrict__ denom,
                                float* __restrict__ out,
                                int nEdges)
{
    __shared__ float sE[128][ADIM];      // 32 KB  tanh(Q[src]+K[dst]) tile
    __shared__ float sWe[ADIM][DIMC];    // 32 KB  We staged once per block
    __shared__ int   sSrc[128];
    __shared__ int   sDst[128];
    __shared__ float sAlpha[128];

    const int tid  = threadIdx.x;
    const int base = blockIdx.x * 128;

    __builtin_prefetch(We, 0, 1);        // global_prefetch_b8: warm L2 for We

    // ---- Stage We (row-major [64][128], contiguous copy of 32 KB) ----
#if USE_ASYNC_LDS
    {
        // Async DMA: 16 B per op, 2048 ops per block, tracked by ASYNCcnt.
        // Builtin signature (from hipcc diagnostic): (v4i __device__*, v4i __shared__*, Ii, Ii)
        AS3 char*       ldsW = (AS3 char*)&sWe[0][0];
        AS1 const char* gW   = (AS1 const char*)We;
        for (int j = tid; j < (ADIM * DIMC) / 4; j += 256) {
            __builtin_amdgcn_global_load_async_to_lds_b128(
                (AS1 v4i*)(gW + (size_t)j * 16),
                (AS3 v4i*)(ldsW + j * 16),
                0, 0);
        }
        wait_asynccnt0();
    }
#else
    for (int i = tid; i < ADIM * DIMC; i += 256)
        sWe[i >> 7][i & 127] = We[i];
#endif

    // Stage per-edge metadata; invalid tail edges get alpha = 0.
    if (tid < 128) {
        int e = base + tid;
        bool valid = (e < nEdges);
        if (!valid) e = nEdges - 1;
        const int s = srcIdx[e];
        const int d = dstIdx[e];
        sSrc[tid] = s;
        sDst[tid] = d;
        sAlpha[tid] = valid ? (exps[e] / (denom[s] + 1e-9f)) : 0.0f;
    }
    __syncthreads();

    // Stage e = tanh(q + k): 8192 elements, 32 per thread.
    for (int i = tid; i < 128 * ADIM; i += 256) {
        const int el = i >> 6;
        const int k  = i & 63;
        sE[el][k] = tanhf(Q[(size_t)sSrc[el] * ADIM + k] +
                          K[(size_t)sDst[el] * ADIM + k]);
    }
    __syncthreads();

    const int wave = tid >> 5;
    const int lane = tid & 31;
    const int m0 = wave * 16;            // this wave's 16 edges (local)
    const int mr = lane & 15;
    const int kh = lane >> 4;

    v8f c[8] = {};                       // 8 N-tiles of 16 cols = 128 output cols
    for (int k0 = 0; k0 < ADIM; k0 += 4) {
        v2f a;
        a.x = sE[m0 + mr][k0 + 2 * kh];
        a.y = sE[m0 + mr][k0 + 2 * kh + 1];
        #pragma unroll
        for (int n = 0; n < 8; ++n) {
            v2f b;
            b.x = sWe[k0 + 2 * kh]    [n * 16 + mr];
            b.y = sWe[k0 + 2 * kh + 1][n * 16 + mr];
            c[n] = __builtin_amdgcn_wmma_f32_16x16x4_f32(false, a, false, b,
                                                         (short)0, c[n],
                                                         false, false);
        }
    }

    // Epilogue: msg = alpha * (Y[dst] + ve), scattered into out[src].
    #pragma unroll
    for (int n = 0; n < 8; ++n) {
        #pragma unroll
        for (int r = 0; r < 8; ++r) {
            const int el  = m0 + r + 8 * kh;         // C layout: M = r + 8*(lane/16)
            const int col = n * 16 + mr;
            const float val = sAlpha[el] *
                (Y[(size_t)sDst[el] * DIMC + col] + c[n][r]);
            atomicAdd(&out[(size_t)sSrc[el] * DIMC + col], val);
        }
    }
}

// ---------------------------------------------------------------------------
// Host-side launch.
// ---------------------------------------------------------------------------
extern "C" void kernel_launch(void* const* d_in, const int* in_sizes, int n_in,
                              void* d_out, int out_size, void* d_ws, size_t ws_size,
                              hipStream_t stream)
{
    const float* X    = (const float*)d_in[0];
    const int*   edge = (const int*)  d_in[1];   // [2, E]
    const float* Wq   = (const float*)d_in[2];
    const float* Wk   = (const float*)d_in[3];
    const float* Wphi = (const float*)d_in[4];
    const float* Wy_w = (const float*)d_in[5];
    const float* Wy_b = (const float*)d_in[6];
    const float* We   = (const float*)d_in[7];
    float*       out  = (float*)d_out;

    const int N = in_sizes[0] / DIMC;            // 50000
    const int E = in_sizes[1] / 2;               // 800000
    const int* src = edge;
    const int* dst = edge + E;

    // Workspace partition (fp32): Q[N,64] K[N,64] Y[N,128] exps[E] denom[N]
    char* ws = (char*)d_ws;
    float* Qb    = (float*)ws; ws += (size_t)N * ADIM * sizeof(float);
    float* Kb    = (float*)ws; ws += (size_t)N * ADIM * sizeof(float);
    float* Yb    = (float*)ws; ws += (size_t)N * DIMC * sizeof(float);
    float* exps  = (float*)ws; ws += (size_t)E * sizeof(float);
    float* denom = (float*)ws;

    // 1) node GEMMs: 16 wave-jobs per 16-row tile, 8 waves per block
    {
        const int nRowTiles = (N + 15) / 16;
        const int jobs = nRowTiles * 16;
        const int blocks = (jobs + 7) / 8;
        node_gemm_kernel<<<blocks, 256, 0, stream>>>(X, Wq, Wk, Wy_w, Wy_b,
                                                     Qb, Kb, Yb, N, nRowTiles);
    }

    // 2) zero accumulators
    {
        const int nOut = N * DIMC;
        const int blocks = (nOut + 255) / 256;
        zero_kernel<<<blocks, 256, 0, stream>>>(out, denom, nOut, N);
    }

    // 3) edge scores + denominator
    {
        const int blocks = (E + 255) / 256;
        edge_score_kernel<<<blocks, 256, 0, stream>>>(src, dst, Qb, Kb, Wphi,
                                                      exps, denom, E);
    }

    // 4) edge message GEMM + scatter
    {
        const int blocks = (E + 127) / 128;
        edge_msg_kernel<<<blocks, 256, 0, stream>>>(src, dst, Qb, Kb, Yb, We,
                                                    exps, denom, out, E);
    }
}